// NonMaximumSuppression_38869454029301
// MI455X (gfx1250) — compile-verified
//
#include <hip/hip_runtime.h>

#ifndef __has_builtin
#define __has_builtin(x) 0
#endif

// TDM path only exists on the device pass for gfx1250; host pass falls back.
#if defined(__gfx1250__) && __has_builtin(__builtin_amdgcn_tensor_load_to_lds) && __has_builtin(__builtin_amdgcn_s_wait_tensorcnt)
#define NMS_USE_TDM 1
#endif

typedef unsigned int u32;
typedef unsigned long long u64;

constexpr int   kB       = 8;
constexpr int   kN       = 2048;
constexpr int   kC       = 21;
constexpr int   kMPC     = 100;   // max per class
constexpr int   kMT      = 200;   // max total
constexpr float kIouT    = 0.5f;
constexpr float kScoreT  = 0.3f;
constexpr float kNeg     = -1.0f;
constexpr int   kThreads = 256;   // 8 wave32s

// Order-preserving float <-> u32 transform, inverted for DESCENDING sort:
// ascending u64 key sort == descending score, ties broken by ascending index.
__device__ __forceinline__ u32 enc_desc(float f) {
  u32 b = __float_as_uint(f);
  u32 u = (b & 0x80000000u) ? ~b : (b | 0x80000000u);  // ascending map
  return ~u;                                           // flip -> descending
}
__device__ __forceinline__ float dec_desc(u32 ku) {
  u32 u = ~ku;
  u32 b = (u & 0x80000000u) ? (u & 0x7fffffffu) : ~u;
  return __uint_as_float(b);
}

#if defined(NMS_USE_TDM)
typedef __attribute__((ext_vector_type(4))) u32 v4u_t;
typedef __attribute__((ext_vector_type(8))) int v8i_t;
typedef __attribute__((ext_vector_type(4))) int v4i_t;

// 1-row 2D tensor copy: nelem f32 elements, global -> LDS, via the TDM.
// Descriptor bitfields per CDNA5 ISA.
__device__ __forceinline__ void tdm_load_f32(const void* gptr, void* lptr, u32 nelem) {
  u64 ga = (u64)(uintptr_t)gptr;
  u32 la = (u32)(uintptr_t)lptr;  // low 32 bits of flat shared ptr = LDS byte offset
  // group0: count=1 | lds_addr | global_addr[31:0] | global_addr[56:32], type=2
  v4u_t g0 = { 1u, la, (u32)ga, (u32)((ga >> 32) & 0x01ffffffu) | (2u << 30) };
  // group1: data_size=4B(code2); tensor_dim0=nelem; tensor_dim1=1;
  //         tile_dim0=nelem; tile_dim1=1; tile_dim2=0; dim0_stride=nelem
  v8i_t g1 = {
    (int)0x00020000u,                    // wg_mask=0, data_size=2 (4B)
    (int)((nelem & 0xffffu) << 16),      // barrier_addr=0 | tensor_dim0[15:0]
    (int)((nelem >> 16) | (1u << 16)),   // tensor_dim0[31:16] | tensor_dim1[15:0]=1
    (int)(nelem << 16),                  // tensor_dim1[31:16]=0 | tile_dim0
    1,                                   // tile_dim1=1 | tile_dim2=0
    (int)nelem,                          // tensor_dim0_stride[31:0]
    0, 0
  };
  v4i_t z4 = {0, 0, 0, 0};
#if __clang_major__ >= 23
  v8i_t z8 = {0, 0, 0, 0, 0, 0, 0, 0};
  __builtin_amdgcn_tensor_load_to_lds(g0, g1, z4, z4, z8, 0);
#else
  __builtin_amdgcn_tensor_load_to_lds(g0, g1, z4, z4, 0);
#endif
}
#endif  // NMS_USE_TDM

// ---------------------------------------------------------------------------
// Kernel 1: one workgroup per (batch, class).
//   - TDM-stage 2048 boxes (32KB) into LDS, overlapped with score load.
//   - Build 64-bit (score,index) keys, LDS bitonic sort (descending score).
//   - Greedy NMS over sorted order with a 2048-bit LDS keep mask,
//     early exit after 100 kept pivots (forward-only suppression means the
//     first 100 kept ARE the per-class top-100).
//   - Emit 100 (score, box) slots (NEG-padded) to workspace.
// ---------------------------------------------------------------------------
__global__ __launch_bounds__(kThreads)
void nms_per_class_kernel(const float* __restrict__ boxes,
                          const float* __restrict__ scores,
                          float* __restrict__ ws_s,
                          float* __restrict__ ws_b) {
  __shared__ u64    s_key[kN];        // 16 KB
  __shared__ float4 s_box[kN];        // 32 KB (original index order)
  __shared__ u32    s_keep[kN / 32];  // 256 B
  __shared__ int    s_kept[kMPC];
  __shared__ int    s_next, s_cnt, s_piv;

  const int tid = threadIdx.x;
  const int b = blockIdx.x / kC;
  const int c = blockIdx.x % kC;

  const float* gbox = boxes + (size_t)b * kN * 4;
  const float* gsc  = scores + (size_t)b * kN * kC + c;

#if defined(NMS_USE_TDM)
  if (tid == 0) tdm_load_f32(gbox, &s_box[0], (u32)(kN * 4));
#else
  __builtin_prefetch(gbox, 0, 1);
  for (int i = tid; i < kN; i += kThreads)
    s_box[i] = reinterpret_cast<const float4*>(gbox)[i];
#endif

  // Score load (stride C) + key build overlaps the TDM transfer.
  __builtin_prefetch(gsc, 0, 1);
  for (int j = tid; j < kN; j += kThreads) {
    float s = gsc[(size_t)j * kC];
    s = (s > kScoreT) ? s : kNeg;
    s_key[j] = ((u64)enc_desc(s) << 32) | (u32)j;
  }

#if defined(NMS_USE_TDM)
  __builtin_amdgcn_s_wait_tensorcnt(0);  // TENSORcnt is per-wave; others pass through
#endif
  __syncthreads();

  // Bitonic sort, ascending key == descending score.
  for (u32 k = 2; k <= (u32)kN; k <<= 1) {
    for (u32 j = k >> 1; j > 0; j >>= 1) {
      for (u32 i = tid; i < (u32)kN; i += kThreads) {
        u32 ixj = i ^ j;
        if (ixj > i) {
          u64 a = s_key[i], v = s_key[ixj];
          bool up = ((i & k) == 0);
          if (up ? (a > v) : (a < v)) { s_key[i] = v; s_key[ixj] = a; }
        }
      }
      __syncthreads();
    }
  }

  // Keep mask: bit j set iff sorted score j passes the threshold.
  for (int w = tid; w < kN / 32; w += kThreads) {
    u32 m = 0;
    for (int t = 0; t < 32; t++) {
      if (dec_desc((u32)(s_key[w * 32 + t] >> 32)) > kScoreT) m |= (1u << t);
    }
    s_keep[w] = m;
  }
  if (tid == 0) { s_next = 0; s_cnt = 0; s_piv = 0; }
  __syncthreads();

  // Greedy suppression, early exit at kMPC kept pivots.
  for (;;) {
    if (tid == 0) {
      int found = -1;
      if (s_cnt < kMPC) {
        int i = s_next;
        while (i < kN) {
          u32 w = s_keep[i >> 5] >> (i & 31);
          if (w) { found = i + __builtin_ctz(w); break; }
          i = (i & ~31) + 32;
        }
      }
      if (found >= 0) { s_kept[s_cnt] = found; s_cnt += 1; s_next = found + 1; }
      s_piv = found;
    }
    __syncthreads();
    const int piv = s_piv;
    if (piv < 0) break;

    const float4 bi = s_box[(u32)s_key[piv]];
    const float areaA = (bi.z - bi.x) * (bi.w - bi.y);
    for (int j = piv + 1 + tid; j < kN; j += kThreads) {
      const u32 w = (u32)j >> 5, bit = 1u << (j & 31);
      if (s_keep[w] & bit) {
        const float4 bj = s_box[(u32)s_key[j]];
        float iy = fmaxf(0.0f, fminf(bi.z, bj.z) - fmaxf(bi.x, bj.x));
        float ix = fmaxf(0.0f, fminf(bi.w, bj.w) - fmaxf(bi.y, bj.y));
        float inter = iy * ix;
        float uni = areaA + (bj.z - bj.x) * (bj.w - bj.y) - inter;
        float iou = (uni > 0.0f) ? (inter / uni) : 0.0f;
        if (iou > kIouT) atomicAnd(&s_keep[w], ~bit);
      }
    }
    __syncthreads();
  }
  __syncthreads();

  // Emit per-class top-100 (NEG-padded; padding is zero-masked in stage 2).
  const int cnt = s_cnt;
  float* os = ws_s + (size_t)blockIdx.x * kMPC;
  float* ob = ws_b + (size_t)blockIdx.x * kMPC * 4;
  for (int t = tid; t < kMPC; t += kThreads) {
    if (t < cnt) {
      const int p = s_kept[t];
      const u64 kk = s_key[p];
      os[t] = dec_desc((u32)(kk >> 32));
      const float4 bx = s_box[(u32)kk];
      ob[t * 4 + 0] = bx.x; ob[t * 4 + 1] = bx.y;
      ob[t * 4 + 2] = bx.z; ob[t * 4 + 3] = bx.w;
    } else {
      os[t] = kNeg;
      ob[t * 4 + 0] = 0.0f; ob[t * 4 + 1] = 0.0f;
      ob[t * 4 + 2] = 0.0f; ob[t * 4 + 3] = 0.0f;
    }
  }
}

// ---------------------------------------------------------------------------
// Kernel 2: one workgroup per batch. Top-200 of the 2100 class survivors via
// a 4096-wide LDS bitonic sort, then emit [class, score, x1, y1, x2, y2] with
// the score>0.3 zero-mask and the y/x swap from the reference.
// ---------------------------------------------------------------------------
__global__ __launch_bounds__(kThreads)
void nms_topk_kernel(const float* __restrict__ ws_s,
                     const float* __restrict__ ws_b,
                     float* __restrict__ out) {
  constexpr int M  = kC * kMPC;  // 2100
  constexpr int NS = 4096;
  __shared__ u64 s_key[NS];      // 32 KB

  const int b = blockIdx.x;
  const int tid = threadIdx.x;
  const float* ss = ws_s + (size_t)b * M;
  const float negInf = __uint_as_float(0xff800000u);

  __builtin_prefetch(ss, 0, 1);
  for (int i = tid; i < NS; i += kThreads) {
    float s = (i < M) ? ss[i] : negInf;
    s_key[i] = ((u64)enc_desc(s) << 32) | (u32)i;
  }
  __syncthreads();

  for (u32 k = 2; k <= (u32)NS; k <<= 1) {
    for (u32 j = k >> 1; j > 0; j >>= 1) {
      for (u32 i = tid; i < (u32)NS; i += kThreads) {
        u32 ixj = i ^ j;
        if (ixj > i) {
          u64 a = s_key[i], v = s_key[ixj];
          bool up = ((i & k) == 0);
          if (up ? (a > v) : (a < v)) { s_key[i] = v; s_key[ixj] = a; }
        }
      }
      __syncthreads();
    }
  }

  for (int t = tid; t < kMT; t += kThreads) {
    const u64 kk = s_key[t];
    const float s = dec_desc((u32)(kk >> 32));
    const int idx = (int)(u32)kk;
    const bool valid = (idx < M) && (s > kScoreT);
    const int gi = valid ? idx : 0;
    const float* bb = ws_b + ((size_t)b * M + gi) * 4;
    const float y0 = bb[0], x0 = bb[1], y1 = bb[2], x1 = bb[3];
    float* o = out + ((size_t)b * kMT + t) * 6;
    o[0] = valid ? (float)(idx / kMPC) : 0.0f;  // class
    o[1] = valid ? s : 0.0f;                    // score
    o[2] = valid ? x0 : 0.0f;                   // xmin
    o[3] = valid ? y0 : 0.0f;                   // ymin
    o[4] = valid ? x1 : 0.0f;                   // xmax
    o[5] = valid ? y1 : 0.0f;                   // ymax
  }
}

extern "C" void kernel_launch(void* const* d_in, const int* in_sizes, int n_in,
                              void* d_out, int out_size, void* d_ws, size_t ws_size,
                              hipStream_t stream) {
  (void)in_sizes; (void)n_in; (void)out_size; (void)ws_size;
  const float* boxes  = (const float*)d_in[0];  // (8, 2048, 4) f32
  const float* scores = (const float*)d_in[1];  // (8, 2048, 21) f32
  float* ws_s = (float*)d_ws;                           // 8*21*100 f32
  float* ws_b = ws_s + (size_t)kB * kC * kMPC;          // 8*21*100*4 f32
  nms_per_class_kernel<<<kB * kC, kThreads, 0, stream>>>(boxes, scores, ws_s, ws_b);
  nms_topk_kernel<<<kB, kThreads, 0, stream>>>(ws_s, ws_b, (float*)d_out);
}